// GCN_17626545783593
// MI455X (gfx1250) — compile-verified
//
#include <hip/hip_runtime.h>
#include <hip/hip_bf16.h>

typedef float v2f __attribute__((ext_vector_type(2)));
typedef float v8f __attribute__((ext_vector_type(8)));

#define HID 128
#define BN_EPS 1e-5f

// ---------------------------------------------------------------------------
// zero fill
// ---------------------------------------------------------------------------
__global__ void zero_f32(float* __restrict__ p, long n) {
  long i = (long)blockIdx.x * blockDim.x + threadIdx.x;
  long stride = (long)gridDim.x * blockDim.x;
  for (; i < n; i += stride) p[i] = 0.0f;
}

// ---------------------------------------------------------------------------
// out[N,HID] = relu(x[N,K] @ W^T + b),  W is [HID,K] row-major.
// One wave -> one 16x16 output tile via V_WMMA_F32_16X16X4_F32.
// Block = 256 threads = 8 waves = all 8 column tiles of HID=128.
// Grid.x = ceil(N/16) row tiles.
// A(m,k): lane = m + 16*(k>>1), vgpr = k&1   (ISA 7.12.2, 32-bit A 16x4)
// B(k,n): lane = n + 16*(k>>1), vgpr = k&1
// D: vgpr v, lane l -> row v + 8*(l>>4), col l&15
// ---------------------------------------------------------------------------
__global__ void linear_relu_wmma(const float* __restrict__ x,
                                 const float* __restrict__ W,
                                 const float* __restrict__ bias,
                                 float* __restrict__ out, int N, int K) {
  const int lane = threadIdx.x & 31;
  const int wave = threadIdx.x >> 5;   // column tile 0..7
  const int row0 = blockIdx.x * 16;
  const int col0 = wave * 16;
  const int t  = lane & 15;            // m for A-loads, n for B-loads/stores
  const int kh = lane >> 4;            // which K half-pair this lane holds

  int rA = row0 + t; if (rA >= N) rA = N - 1;   // clamp loads, keep EXEC full
  const float* __restrict__ xrow = x + (long)rA * K;
  const float* __restrict__ wrow = W + (long)(col0 + t) * K;

  v8f acc = {};
  for (int kk = 0; kk < K; kk += 4) {
    const int ka = kk + 2 * kh;
    v2f a, b;
    a.x = xrow[ka];  a.y = xrow[ka + 1];
    b.x = wrow[ka];  b.y = wrow[ka + 1];
    acc = __builtin_amdgcn_wmma_f32_16x16x4_f32(false, a, false, b,
                                                (short)0, acc, false, false);
  }

  const float bi = bias[col0 + t];
#pragma unroll
  for (int v = 0; v < 8; ++v) {
    const int row = row0 + v + 8 * kh;
    float val = acc[v] + bi;
    val = val > 0.0f ? val : 0.0f;
    if (row < N) out[(long)row * HID + col0 + t] = val;
  }
}

// ---------------------------------------------------------------------------
// out = agg @ Wrel^T + b + h @ Wroot^T   (both K=HID GEMMs fused per tile)
// ---------------------------------------------------------------------------
__global__ void conv_combine_wmma(const float* __restrict__ agg,
                                  const float* __restrict__ h,
                                  const float* __restrict__ Wrel,
                                  const float* __restrict__ Wroot,
                                  const float* __restrict__ bias,
                                  float* __restrict__ out, int N) {
  const int lane = threadIdx.x & 31;
  const int wave = threadIdx.x >> 5;
  const int row0 = blockIdx.x * 16;
  const int col0 = wave * 16;
  const int t  = lane & 15;
  const int kh = lane >> 4;

  int rA = row0 + t; if (rA >= N) rA = N - 1;
  const float* __restrict__ arow = agg + (long)rA * HID;
  const float* __restrict__ hrow = h   + (long)rA * HID;
  const float* __restrict__ w1 = Wrel  + (long)(col0 + t) * HID;
  const float* __restrict__ w2 = Wroot + (long)(col0 + t) * HID;

  v8f acc = {};
#pragma unroll 4
  for (int kk = 0; kk < HID; kk += 4) {
    const int ka = kk + 2 * kh;
    v2f a, b;
    a.x = arow[ka]; a.y = arow[ka + 1];
    b.x = w1[ka];   b.y = w1[ka + 1];
    acc = __builtin_amdgcn_wmma_f32_16x16x4_f32(false, a, false, b,
                                                (short)0, acc, false, false);
    a.x = hrow[ka]; a.y = hrow[ka + 1];
    b.x = w2[ka];   b.y = w2[ka + 1];
    acc = __builtin_amdgcn_wmma_f32_16x16x4_f32(false, a, false, b,
                                                (short)0, acc, false, false);
  }

  const float bi = bias[col0 + t];
#pragma unroll
  for (int v = 0; v < 8; ++v) {
    const int row = row0 + v + 8 * kh;
    if (row < N) out[(long)row * HID + col0 + t] = acc[v] + bi;
  }
}

// ---------------------------------------------------------------------------
// agg[dst[e]] += ew[e] * h[src[e]]   — wave-per-edge, float4 gather,
// scatter via non-returning global_atomic_add_f32 (L2 atomic units;
// h [25.6 MB] is L2-resident so this stays on-chip).
// ---------------------------------------------------------------------------
__global__ void edge_scatter(const int* __restrict__ src,
                             const int* __restrict__ dst,
                             const float* __restrict__ ew,
                             const float* __restrict__ h,
                             float* __restrict__ agg, int E) {
  const int e = blockIdx.x * (blockDim.x >> 5) + (threadIdx.x >> 5);
  if (e >= E) return;
  const int lane = threadIdx.x & 31;
  const int s = src[e];
  const int d = dst[e];
  const float w = ew[e];
  if (e + 8 < E) {  // prefetch the next batch-of-waves' source row
    __builtin_prefetch(h + (long)src[e + 8] * HID, 0, 1);
  }
  const float4 v = ((const float4*)(h + (long)s * HID))[lane];
  float* ap = agg + (long)d * HID + lane * 4;
  atomicAdd(ap + 0, v.x * w);
  atomicAdd(ap + 1, v.y * w);
  atomicAdd(ap + 2, v.z * w);
  atomicAdd(ap + 3, v.w * w);
}

// ---------------------------------------------------------------------------
// Batch-norm pass 1: per-column sum / sumsq (population stats, matches jnp.var)
// blockDim = 128 (one column per thread), rows strided across grid.
// ---------------------------------------------------------------------------
__global__ void bn_stats(const float* __restrict__ h, float* __restrict__ stats,
                         int N) {
  const int col = threadIdx.x;
  float s = 0.0f, s2 = 0.0f;
  for (int r = blockIdx.x; r < N; r += gridDim.x) {
    const float v = h[(long)r * HID + col];
    s += v;
    s2 += v * v;
  }
  atomicAdd(&stats[col], s);
  atomicAdd(&stats[HID + col], s2);
}

// ---------------------------------------------------------------------------
// Batch-norm pass 2: normalize (+optional relu)
// ---------------------------------------------------------------------------
__global__ void bn_apply(const float* __restrict__ in, float* __restrict__ out,
                         const float* __restrict__ stats,
                         const float* __restrict__ gamma,
                         const float* __restrict__ beta, int N, int do_relu) {
  const long n = (long)N * HID;
  long i = (long)blockIdx.x * blockDim.x + threadIdx.x;
  const long stride = (long)gridDim.x * blockDim.x;
  const float invN = 1.0f / (float)N;
  for (; i < n; i += stride) {
    const int col = (int)(i & (HID - 1));
    const float mu = stats[col] * invN;
    const float var = stats[HID + col] * invN - mu * mu;
    float v = (in[i] - mu) * rsqrtf(var + BN_EPS) * gamma[col] + beta[col];
    if (do_relu) v = v > 0.0f ? v : 0.0f;
    out[i] = v;
  }
}

// ---------------------------------------------------------------------------
extern "C" void kernel_launch(void* const* d_in, const int* in_sizes, int n_in,
                              void* d_out, int out_size, void* d_ws,
                              size_t ws_size, hipStream_t stream) {
  const float* x       = (const float*)d_in[0];
  const int*   adj     = (const int*)d_in[1];
  const float* feat    = (const float*)d_in[2];
  const float* W_in    = (const float*)d_in[3];
  const float* b_in    = (const float*)d_in[4];
  const float* W1_rel  = (const float*)d_in[5];
  const float* b1_rel  = (const float*)d_in[6];
  const float* W1_root = (const float*)d_in[7];
  const float* W2_rel  = (const float*)d_in[8];
  const float* b2_rel  = (const float*)d_in[9];
  const float* W2_root = (const float*)d_in[10];
  const float* gamma   = (const float*)d_in[11];
  const float* beta    = (const float*)d_in[12];

  const int nhid  = in_sizes[4];           // 128
  const int nfeat = in_sizes[3] / nhid;    // 256
  const int N     = in_sizes[0] / nfeat;   // 50000
  const int E     = in_sizes[2];           // 800000
  const int* src = adj;
  const int* dstp = adj + E;

  float* hA    = (float*)d_ws;             // [N, HID]
  float* agg   = hA  + (long)N * HID;      // [N, HID]
  float* hB    = agg + (long)N * HID;      // [N, HID]
  float* stats = hB  + (long)N * HID;      // [2*HID]

  const long nh = (long)N * HID;
  const int rowTiles = (N + 15) / 16;
  const int zeroBlocks = (int)((nh + 255) / 256);
  const int edgeBlocks = (E + 7) / 8;      // 8 waves per 256-thread block

  // h = relu(x @ W_in^T + b_in)
  linear_relu_wmma<<<rowTiles, 256, 0, stream>>>(x, W_in, b_in, hA, N, nfeat);

  // ---- conv 1 ----
  zero_f32<<<zeroBlocks, 256, 0, stream>>>(agg, nh);
  zero_f32<<<1, 256, 0, stream>>>(stats, 2 * HID);
  edge_scatter<<<edgeBlocks, 256, 0, stream>>>(src, dstp, feat, hA, agg, E);
  conv_combine_wmma<<<rowTiles, 256, 0, stream>>>(agg, hA, W1_rel, W1_root,
                                                  b1_rel, hB, N);
  // relu(batch_norm(hB)) -> hA
  bn_stats<<<512, HID, 0, stream>>>(hB, stats, N);
  bn_apply<<<2048, 256, 0, stream>>>(hB, hA, stats, gamma, beta, N, 1);

  // ---- conv 2 ----
  zero_f32<<<zeroBlocks, 256, 0, stream>>>(agg, nh);
  zero_f32<<<1, 256, 0, stream>>>(stats, 2 * HID);
  edge_scatter<<<edgeBlocks, 256, 0, stream>>>(src, dstp, feat, hA, agg, E);
  conv_combine_wmma<<<rowTiles, 256, 0, stream>>>(agg, hA, W2_rel, W2_root,
                                                  b2_rel, hB, N);
  // batch_norm(hB) -> d_out
  bn_stats<<<512, HID, 0, stream>>>(hB, stats, N);
  bn_apply<<<2048, 256, 0, stream>>>(hB, (float*)d_out, stats, gamma, beta, N, 0);
}